// ComputeFlowFields_12867722019020
// MI455X (gfx1250) — compile-verified
//
#include <hip/hip_runtime.h>

typedef __attribute__((ext_vector_type(2))) float v2f;
typedef __attribute__((ext_vector_type(8))) float v8f;

constexpr int Bn = 4, Dn = 32, Hn = 384, Wn = 512;
constexpr int HWn = Hn * Wn;

__global__ __launch_bounds__(256) void
flowfields_wmma(const float* __restrict__ hm,
                const float* __restrict__ alpha,
                float* __restrict__ out)
{
  const int lane = threadIdx.x & 31;
  const int wave = threadIdx.x >> 5;
  const int b    = blockIdx.y;
  const int tile = blockIdx.x * 8 + wave;   // 16-pixel tile within image
  const int px0  = tile * 16;
  const int l15  = lane & 15;
  const bool hi  = lane >= 16;

  const float fy  = (float)(px0 >> 9);        // W = 512
  const float fx0 = (float)(px0 & (Wn - 1));

  // ---- B matrix (4x16): column n = [x_n, y, 1, 0]^T
  // lanes 0-15: VGPR0=K0 (x), VGPR1=K1 (y); lanes 16-31: VGPR0=K2 (1), VGPR1=K3 (0)
  v2f bm;
  bm.x = hi ? 1.0f : (fx0 + (float)l15);
  bm.y = hi ? 0.0f : fy;

  const float invhw = 1.0f / ((Wn - 1) * 0.5f);   // fold [-1,1] normalization
  const float invhh = 1.0f / ((Hn - 1) * 0.5f);   // into the homography rows

  // ---- A matrices (16x4): set s covers planes d = 16*s + row
  // lanes 0-15 hold K=0,1 of row M=lane; lanes 16-31 hold K=2,3 of row M=lane-16
  v2f ax[2], ay[2], az[2];
  #pragma unroll
  for (int s = 0; s < 2; ++s) {
    const float* m = hm + (size_t)((b * Dn) + 16 * s + l15) * 9;
    float r0 = m[0], r1 = m[1], r2 = m[2];
    float r3 = m[3], r4 = m[4], r5 = m[5];
    float r6 = m[6], r7 = m[7], r8 = m[8];
    ax[s].x = (hi ? r2 : r0) * invhw;  ax[s].y = hi ? 0.0f : r1 * invhw;
    ay[s].x = (hi ? r5 : r3) * invhh;  ay[s].y = hi ? 0.0f : r4 * invhh;
    az[s].x =  hi ? r8 : r6;           az[s].y = hi ? 0.0f : r7;
  }

  // ---- Stream alphas (coalesced, all issued early for MLP)
  // lane handles d = dbase+v (pass A) and d = dbase+16+v (pass B)
  const int dbase = hi ? 8 : 0;
  const float* ap = alpha + (size_t)(b * Dn + dbase) * HWn + px0 + l15;
  float aA[8], aB[8];
  #pragma unroll
  for (int v = 0; v < 8; ++v) aA[v] = ap[(size_t)v * HWn];
  #pragma unroll
  for (int v = 0; v < 8; ++v) aB[v] = ap[(size_t)(16 + v) * HWn];

  // ---- Homography via WMMA: D[d, pixel] with lane = pixel, VGPR index = d
  v8f cz = {};
  v8f wx0 = __builtin_amdgcn_wmma_f32_16x16x4_f32(false, ax[0], false, bm, (short)0, cz, false, false);
  v8f wy0 = __builtin_amdgcn_wmma_f32_16x16x4_f32(false, ay[0], false, bm, (short)0, cz, false, false);
  v8f wz0 = __builtin_amdgcn_wmma_f32_16x16x4_f32(false, az[0], false, bm, (short)0, cz, false, false);
  v8f wx1 = __builtin_amdgcn_wmma_f32_16x16x4_f32(false, ax[1], false, bm, (short)0, cz, false, false);
  v8f wy1 = __builtin_amdgcn_wmma_f32_16x16x4_f32(false, ay[1], false, bm, (short)0, cz, false, false);
  v8f wz1 = __builtin_amdgcn_wmma_f32_16x16x4_f32(false, az[1], false, bm, (short)0, cz, false, false);

  // ---- Pass A: composite this lane's first 8-deep segment (d ascending in-lane)
  float T = 1.0f, accx = 0.0f, accy = 0.0f;
  #pragma unroll
  for (int v = 0; v < 8; ++v) {
    float rz = __builtin_amdgcn_rcpf(wz0[v]);
    float xn = __builtin_fmaf(wx0[v], rz, -1.0f);
    float yn = __builtin_fmaf(wy0[v], rz, -1.0f);
    float a   = aA[v];
    float wgt = a * T;
    accx = __builtin_fmaf(xn, wgt, accx);
    accy = __builtin_fmaf(yn, wgt, accy);
    T    = __builtin_fmaf(-a, T, T);          // T *= (1 - a)
  }

  // ---- Pass B: second segment (d += 16), independent local transmittance
  float T2 = 1.0f, accx2 = 0.0f, accy2 = 0.0f;
  #pragma unroll
  for (int v = 0; v < 8; ++v) {
    float rz = __builtin_amdgcn_rcpf(wz1[v]);
    float xn = __builtin_fmaf(wx1[v], rz, -1.0f);
    float yn = __builtin_fmaf(wy1[v], rz, -1.0f);
    float a   = aB[v];
    float wgt = a * T2;
    accx2 = __builtin_fmaf(xn, wgt, accx2);
    accy2 = __builtin_fmaf(yn, wgt, accy2);
    T2    = __builtin_fmaf(-a, T2, T2);
  }

  // ---- Stitch exclusive cumprod across the lane pair (lane <-> lane^16):
  // out = a0 + P0*(a1 + P1*(a2 + P2*a3))
  // lane<16 holds (a0,P0)=d0-7, (a2,P2)=d16-23; lane>=16 holds (a1,P1)=d8-15, a3=d24-31
  float t1 = __shfl_xor(accx,  16, 32);
  float t2 = __shfl_xor(accy,  16, 32);
  float t3 = __shfl_xor(T,     16, 32);
  float t4 = __shfl_xor(accx2, 16, 32);
  float t5 = __shfl_xor(accy2, 16, 32);

  if (!hi) {
    float ox = __builtin_fmaf(T, t1 + t3 * __builtin_fmaf(T2, t4, accx2), accx);
    float oy = __builtin_fmaf(T, t2 + t3 * __builtin_fmaf(T2, t5, accy2), accy);
    float2 o; o.x = ox; o.y = oy;
    *(float2*)(out + (size_t)(b * HWn + px0 + l15) * 2) = o;   // (B,H,W,2)
  }
}

extern "C" void kernel_launch(void* const* d_in, const int* in_sizes, int n_in,
                              void* d_out, int out_size, void* d_ws, size_t ws_size,
                              hipStream_t stream) {
  const float* hm    = (const float*)d_in[0];   // (B, D, 3, 3) fp32
  const float* alpha = (const float*)d_in[1];   // (B, D, 1, H, W) fp32
  float* out         = (float*)d_out;           // (B, H, W, 2) fp32

  dim3 grid(HWn / (16 * 8), Bn);   // 8 waves/block, one 16-pixel tile per wave
  dim3 block(256);
  flowfields_wmma<<<grid, block, 0, stream>>>(hm, alpha, out);
}